// Head_13941463842981
// MI455X (gfx1250) — compile-verified
//
#include <hip/hip_runtime.h>

// Types for the f32 WMMA: A is 16x4 f32 (2 VGPRs), B is 4x16 f32 (2 VGPRs),
// C/D is 16x16 f32 (8 VGPRs).
typedef __attribute__((ext_vector_type(2))) float v2f;
typedef __attribute__((ext_vector_type(8))) float v8f;

#define SEQ_T 1024

// One block per batch row. 128 threads = 4 wave32s; wave w owns elements
// [w*256, w*256+256). Computes, per batch b:
//   s_j = c*x_j^2, M = max_j s_j, e_j = exp(s_j - M), f_j = e_j * wv * x_j
//   E_j, F_j = inclusive prefix sums of e, f
//   out[b,0] = 0 ; out[b,i] = F_{i-1} / E_{i-1}
// Intra-wave 256-scan is done as Y = L @ V (L = 16x16 lower-tri ones) via a
// K-split chain of v_wmma_f32_16x16x4_f32; cross-segment offsets via a 16-lane
// shuffle scan of column totals; cross-wave offsets via LDS.
__global__ __launch_bounds__(128) void head_scan_kernel(
    const float* __restrict__ X, const float* __restrict__ Wk,
    const float* __restrict__ Wq, const float* __restrict__ Wv,
    float* __restrict__ out, int Bn) {
  const int b = blockIdx.x;
  if (b >= Bn) return;
  const int tid  = threadIdx.x;
  const int wave = tid >> 5;
  const int lane = tid & 31;
  const int h = lane >> 4;   // lane half (selects K/M sub-range per ISA layout)
  const int n = lane & 15;   // column / A-row index

  const float c1 = Wq[0] * Wk[0];
  const float wv = Wv[0];
  const float* xb = X + (size_t)b * SEQ_T;
  float* ob = out + (size_t)b * SEQ_T;

  __shared__ float sM[4];
  __shared__ float sE[4];
  __shared__ float sF[4];

  // ---- Phase 1: global max of s = c1*x^2 over this batch row --------------
  float m = -3.402823466e+38f;
  for (int k = tid; k < SEQ_T; k += 128) {
    float x = xb[k];
    float s = c1 * x * x;
    m = fmaxf(m, s);
  }
  for (int d = 16; d >= 1; d >>= 1) m = fmaxf(m, __shfl_xor(m, d, 32));
  if (lane == 0) sM[wave] = m;
  __syncthreads();
  m = fmaxf(fmaxf(sM[0], sM[1]), fmaxf(sM[2], sM[3]));

  // ---- Phase 2: intra-wave 256-element tile scan via WMMA -----------------
  // Tile element (r, n) <-> chunk index n*16 + r (column-major 16-segments).
  const int base = wave * 256;
  v8f yE = {};  // E inclusive intra-column prefixes (C/D layout)
  v8f yF = {};  // F inclusive intra-column prefixes
#pragma unroll
  for (int q = 0; q < 4; ++q) {
    v2f a, bE, bF;
#pragma unroll
    for (int j = 0; j < 2; ++j) {
      const int r = 4 * q + 2 * h + j;        // K index (A) / row index (B)
      const float x = xb[base + n * 16 + r];  // B-layout load
      const float s = c1 * x * x;
      const float e = __expf(s - m);          // in (0, 1]
      a[j]  = (r <= n) ? 1.0f : 0.0f;         // L[m=n, k=r] lower-triangular
      bE[j] = e;
      bF[j] = e * wv * x;
    }
    yE = __builtin_amdgcn_wmma_f32_16x16x4_f32(false, a, false, bE,
                                               (short)0, yE, false, false);
    yF = __builtin_amdgcn_wmma_f32_16x16x4_f32(false, a, false, bF,
                                               (short)0, yF, false, false);
  }

  // Column totals live in Y[15, n] = C VGPR 7, upper lane half. Broadcast the
  // 16 totals to both halves, shuffle-scan across the 16 columns.
  float tE = __shfl(yE[7], 16 + n, 32);
  float tF = __shfl(yF[7], 16 + n, 32);
  float iE = tE, iF = tF;
#pragma unroll
  for (int d = 1; d < 16; d <<= 1) {
    float uE = __shfl_up(iE, d, 16);
    float uF = __shfl_up(iF, d, 16);
    if (n >= d) { iE += uE; iF += uF; }
  }
  float offE = iE - tE;  // exclusive cross-column offset within chunk
  float offF = iF - tF;
  float chunkE = __shfl(iE, 15, 16);  // whole-chunk sums (uniform per wave)
  float chunkF = __shfl(iF, 15, 16);

  // ---- Phase 3: cross-wave (cross-chunk) offsets via LDS ------------------
  if (lane == 0) { sE[wave] = chunkE; sF[wave] = chunkF; }
  __syncthreads();
  for (int w = 0; w < wave; ++w) { offE += sE[w]; offF += sF[w]; }

  // ---- Emit: out[idx+1] = F_idx / E_idx ; out[0] = 0 ----------------------
  // C/D layout: yE[d] on this lane = Y[d + 8h, n] -> chunk idx n*16 + d + 8h.
#pragma unroll
  for (int d = 0; d < 8; ++d) {
    const int idx = base + n * 16 + (d + 8 * h);
    const float E = yE[d] + offE;
    const float F = yF[d] + offF;
    if (idx + 1 < SEQ_T) ob[idx + 1] = F / E;
  }
  if (tid == 0) ob[0] = 0.0f;
}

extern "C" void kernel_launch(void* const* d_in, const int* in_sizes, int n_in,
                              void* d_out, int out_size, void* d_ws,
                              size_t ws_size, hipStream_t stream) {
  const float* X  = (const float*)d_in[0];
  const float* Wk = (const float*)d_in[1];
  const float* Wq = (const float*)d_in[2];
  const float* Wv = (const float*)d_in[3];
  float* out = (float*)d_out;
  const int Bn = in_sizes[0] / SEQ_T;  // 128 for the reference shapes
  head_scan_kernel<<<dim3(Bn), dim3(128), 0, stream>>>(X, Wk, Wq, Wv, out, Bn);
}